// GNNLinkPredictor_11338713662168
// MI455X (gfx1250) — compile-verified
//
#include <hip/hip_runtime.h>
#include <math.h>

typedef __attribute__((ext_vector_type(2))) float v2f;
typedef __attribute__((ext_vector_type(8))) float v8f;

// fp32 WMMA: D = A(16x4) x B(4x16) + C(16x16), exact fp32 (RNE) per CDNA5 ISA.
#define WMMA_F32(a, b, c) \
  __builtin_amdgcn_wmma_f32_16x16x4_f32(false, (a), false, (b), (short)0, (c), false, false)

// ---------------------------------------------------------------------------
// Utility: grid-stride zero fill (graph-capture safe; no hipMemset)
// ---------------------------------------------------------------------------
__global__ void zero_f32(float* __restrict__ p, size_t n) {
  size_t i = (size_t)blockIdx.x * blockDim.x + threadIdx.x;
  size_t stride = (size_t)gridDim.x * blockDim.x;
  for (; i < n; i += stride) p[i] = 0.0f;
}

// ---------------------------------------------------------------------------
// Edge degree counts (float, to match reference's segment_sum of ones)
// ---------------------------------------------------------------------------
__global__ void count_edges(const int* __restrict__ dst, float* __restrict__ cnt, int E) {
  int e = blockIdx.x * blockDim.x + threadIdx.x;
  if (e < E) atomicAdd(&cnt[dst[e]], 1.0f);
}

// ---------------------------------------------------------------------------
// Scatter-add of source node features into destination accumulators.
// One wave (32 lanes) per edge; lanes cover consecutive features (coalesced).
// ---------------------------------------------------------------------------
__global__ void scatter_add(const float* __restrict__ x, const int* __restrict__ src,
                            const int* __restrict__ dst, float* __restrict__ agg,
                            int E, int D) {
  int tid = blockIdx.x * blockDim.x + threadIdx.x;
  int e = tid >> 5;
  int lane = tid & 31;
  if (e >= E) return;
  const float* xs = x + (size_t)src[e] * D;
  float* ad = agg + (size_t)dst[e] * D;
  for (int f = lane; f < D; f += 32) atomicAdd(&ad[f], xs[f]);
}

// ---------------------------------------------------------------------------
// SAGE linear: out[m, 0:128] = opt_relu( (agg[m,:K1]/max(cnt[m],1)) @ Wl^T
//                                        + bl + xroot[m,:K2] @ Wr^T )
// One wave per 16-row tile; 8 WMMA accumulators cover all 128 output columns.
// Grid = N/16 (N divisible by 16 for all call sites), blockDim = 32 (wave32,
// EXEC all ones as WMMA requires).
// ---------------------------------------------------------------------------
__global__ __launch_bounds__(32)
void sage_gemm(const float* __restrict__ agg, const float* __restrict__ cnt,
               const float* __restrict__ Wl, const float* __restrict__ bl,
               const float* __restrict__ xroot, const float* __restrict__ Wr,
               float* __restrict__ out, int K1, int K2, int do_relu) {
  const int lane  = threadIdx.x;
  const int m0    = blockIdx.x * 16;
  const int mrow  = m0 + (lane & 15);         // A-fragment row for this lane
  const int khalf = (lane >> 4) << 1;         // lanes 0-15 -> K+0/+1, 16-31 -> K+2/+3
  const int col   = lane & 15;                // B/D column within a 16-wide tile

  v8f acc[8];
#pragma unroll
  for (int t = 0; t < 8; ++t) acc[t] = (v8f){0, 0, 0, 0, 0, 0, 0, 0};

  // ---- neighbor-mean part: (agg/cnt) @ Wl^T ; fold 1/cnt into A loads ----
  {
    const float rinv = 1.0f / fmaxf(cnt[mrow], 1.0f);
    const float* Arow = agg + (size_t)mrow * K1;
    for (int k0 = 0; k0 < K1; k0 += 4) {
      v2f a;
      a.x = Arow[k0 + khalf] * rinv;
      a.y = Arow[k0 + khalf + 1] * rinv;
#pragma unroll
      for (int t = 0; t < 8; ++t) {
        const float* Wp = Wl + (size_t)(t * 16 + col) * K1 + k0 + khalf;
        v2f b; b.x = Wp[0]; b.y = Wp[1];
        acc[t] = WMMA_F32(a, b, acc[t]);
      }
    }
  }
  // ---- root part: xroot @ Wr^T ----
  {
    const float* Arow = xroot + (size_t)mrow * K2;
    for (int k0 = 0; k0 < K2; k0 += 4) {
      v2f a;
      a.x = Arow[k0 + khalf];
      a.y = Arow[k0 + khalf + 1];
#pragma unroll
      for (int t = 0; t < 8; ++t) {
        const float* Wp = Wr + (size_t)(t * 16 + col) * K2 + k0 + khalf;
        v2f b; b.x = Wp[0]; b.y = Wp[1];
        acc[t] = WMMA_F32(a, b, acc[t]);
      }
    }
  }
  // ---- writeback (+bias, optional relu). D layout: VGPR j -> rows m0+j / m0+j+8 ----
  const int rbase = m0 + ((lane >> 4) << 3);
#pragma unroll
  for (int t = 0; t < 8; ++t) {
    const int n = t * 16 + col;
    const float bias = bl[n];
#pragma unroll
    for (int j = 0; j < 8; ++j) {
      float v = acc[t][j] + bias;
      if (do_relu) v = fmaxf(v, 0.0f);
      out[(size_t)(rbase + j) * 128 + n] = v;
    }
  }
}

// ---------------------------------------------------------------------------
// Fused link predictor:
//   x = [z_inv[eli_inv] | z_fund[eli_fund]]  (EL x 256)
//   h = relu(x @ Wp1^T + bp1)                (EL x 128, kept in accumulators)
//   out = sigmoid(h @ Wp2^T + bp2)           (EL x 1)
// One wave per 16-row tile. Second layer done via per-lane partials +
// shfl_xor reduction inside each 16-lane half (matches C/D VGPR layout).
// ---------------------------------------------------------------------------
__global__ __launch_bounds__(32)
void link_predictor(const float* __restrict__ z_inv, const float* __restrict__ z_fund,
                    const int* __restrict__ eli_inv, const int* __restrict__ eli_fund,
                    const float* __restrict__ Wp1, const float* __restrict__ bp1,
                    const float* __restrict__ Wp2, const float* __restrict__ bp2,
                    float* __restrict__ out) {
  const int lane  = threadIdx.x;
  const int r0    = blockIdx.x * 16;
  const int m     = r0 + (lane & 15);
  const int khalf = (lane >> 4) << 1;
  const int col   = lane & 15;

  const float* zi = z_inv + (size_t)eli_inv[m] * 128;
  const float* zj = z_fund + (size_t)eli_fund[m] * 128;

  v8f acc[8];
#pragma unroll
  for (int t = 0; t < 8; ++t) acc[t] = (v8f){0, 0, 0, 0, 0, 0, 0, 0};

  // K = 0..127 from z_inv gather, Wp1 columns [0,128)
  for (int k0 = 0; k0 < 128; k0 += 4) {
    v2f a;
    a.x = zi[k0 + khalf];
    a.y = zi[k0 + khalf + 1];
#pragma unroll
    for (int t = 0; t < 8; ++t) {
      const float* Wp = Wp1 + (size_t)(t * 16 + col) * 256 + k0 + khalf;
      v2f b; b.x = Wp[0]; b.y = Wp[1];
      acc[t] = WMMA_F32(a, b, acc[t]);
    }
  }
  // K = 128..255 from z_fund gather, Wp1 columns [128,256)
  for (int k0 = 0; k0 < 128; k0 += 4) {
    v2f a;
    a.x = zj[k0 + khalf];
    a.y = zj[k0 + khalf + 1];
#pragma unroll
    for (int t = 0; t < 8; ++t) {
      const float* Wp = Wp1 + (size_t)(t * 16 + col) * 256 + 128 + k0 + khalf;
      v2f b; b.x = Wp[0]; b.y = Wp[1];
      acc[t] = WMMA_F32(a, b, acc[t]);
    }
  }

  // Layer 2: per-lane partial dot of relu(h+bias) with Wp2 over this lane's columns.
  float partial[8];
#pragma unroll
  for (int j = 0; j < 8; ++j) partial[j] = 0.0f;
#pragma unroll
  for (int t = 0; t < 8; ++t) {
    const int n = t * 16 + col;
    const float b1 = bp1[n];
    const float w2 = Wp2[n];
#pragma unroll
    for (int j = 0; j < 8; ++j) {
      float h = fmaxf(acc[t][j] + b1, 0.0f);
      partial[j] += h * w2;
    }
  }
  // Reduce across the 16 lanes of each half (rows j for lanes 0-15, j+8 for 16-31).
#pragma unroll
  for (int off = 1; off < 16; off <<= 1) {
#pragma unroll
    for (int j = 0; j < 8; ++j)
      partial[j] += __shfl_xor(partial[j], off, 32);
  }
  if (col == 0) {
    const int mb = r0 + ((lane >> 4) << 3);
    const float b2 = bp2[0];
#pragma unroll
    for (int j = 0; j < 8; ++j) {
      float s = partial[j] + b2;
      out[mb + j] = 1.0f / (1.0f + expf(-s));
    }
  }
}

// ---------------------------------------------------------------------------
// Host-side launcher (graph-capture safe: kernels only, all on `stream`)
// ---------------------------------------------------------------------------
extern "C" void kernel_launch(void* const* d_in, const int* in_sizes, int n_in,
                              void* d_out, int out_size, void* d_ws, size_t ws_size,
                              hipStream_t stream) {
  (void)in_sizes; (void)n_in; (void)out_size; (void)ws_size;

  const float* x_inv  = (const float*)d_in[0];
  const float* x_fund = (const float*)d_in[1];
  const int* src_if   = (const int*)d_in[2];
  const int* dst_if   = (const int*)d_in[3];
  const int* src_fi   = (const int*)d_in[4];
  const int* dst_fi   = (const int*)d_in[5];
  const int* eli_inv  = (const int*)d_in[6];
  const int* eli_fund = (const int*)d_in[7];
  const float* W1l_if = (const float*)d_in[8];
  const float* b1_if  = (const float*)d_in[9];
  const float* W1r_if = (const float*)d_in[10];
  const float* W1l_fi = (const float*)d_in[11];
  const float* b1_fi  = (const float*)d_in[12];
  const float* W1r_fi = (const float*)d_in[13];
  const float* W2l_if = (const float*)d_in[14];
  const float* b2_if  = (const float*)d_in[15];
  const float* W2r_if = (const float*)d_in[16];
  const float* W2l_fi = (const float*)d_in[17];
  const float* b2_fi  = (const float*)d_in[18];
  const float* W2r_fi = (const float*)d_in[19];
  const float* Wp1    = (const float*)d_in[20];
  const float* bp1    = (const float*)d_in[21];
  const float* Wp2    = (const float*)d_in[22];
  const float* bp2    = (const float*)d_in[23];

  constexpr int N_INV = 100000, N_FUND = 20000, E = 600000, EL = 200000, H = 128;

  // Workspace layout (floats): [agg_if][agg_fi][cnt_f][cnt_i][h_fund][h_inv][z_fund][z_inv]
  float* ws     = (float*)d_ws;
  float* agg_if = ws;                                   // N_FUND x 128
  float* agg_fi = agg_if + (size_t)N_FUND * H;          // N_INV  x 128 (layer1 uses ld=64)
  float* cnt_f  = agg_fi + (size_t)N_INV * H;           // N_FUND
  float* cnt_i  = cnt_f + N_FUND;                       // N_INV
  float* h_fund = cnt_i + N_INV;                        // N_FUND x 128
  float* h_inv  = h_fund + (size_t)N_FUND * H;          // N_INV  x 128
  float* z_fund = h_inv + (size_t)N_INV * H;            // N_FUND x 128
  float* z_inv  = z_fund + (size_t)N_FUND * H;          // N_INV  x 128

  const size_t aggFloats  = (size_t)N_FUND * H + (size_t)N_INV * H;
  const size_t zero1Count = aggFloats + N_FUND + N_INV; // agg buffers + counts

  const int edgeGrid = (E * 32 + 255) / 256;

  // ---- layer 1 ----
  zero_f32<<<4096, 256, 0, stream>>>(ws, zero1Count);
  count_edges<<<(E + 255) / 256, 256, 0, stream>>>(dst_if, cnt_f, E);
  count_edges<<<(E + 255) / 256, 256, 0, stream>>>(dst_fi, cnt_i, E);
  scatter_add<<<edgeGrid, 256, 0, stream>>>(x_inv, src_if, dst_if, agg_if, E, 128);
  scatter_add<<<edgeGrid, 256, 0, stream>>>(x_fund, src_fi, dst_fi, agg_fi, E, 64);
  sage_gemm<<<N_FUND / 16, 32, 0, stream>>>(agg_if, cnt_f, W1l_if, b1_if,
                                            x_fund, W1r_if, h_fund, 128, 64, 1);
  sage_gemm<<<N_INV / 16, 32, 0, stream>>>(agg_fi, cnt_i, W1l_fi, b1_fi,
                                           x_inv, W1r_fi, h_inv, 64, 128, 1);

  // ---- layer 2 (reuse agg buffers; counts unchanged) ----
  zero_f32<<<4096, 256, 0, stream>>>(ws, aggFloats);
  scatter_add<<<edgeGrid, 256, 0, stream>>>(h_inv, src_if, dst_if, agg_if, E, 128);
  scatter_add<<<edgeGrid, 256, 0, stream>>>(h_fund, src_fi, dst_fi, agg_fi, E, 128);
  sage_gemm<<<N_FUND / 16, 32, 0, stream>>>(agg_if, cnt_f, W2l_if, b2_if,
                                            h_fund, W2r_if, z_fund, 128, 128, 0);
  sage_gemm<<<N_INV / 16, 32, 0, stream>>>(agg_fi, cnt_i, W2l_fi, b2_fi,
                                           h_inv, W2r_fi, z_inv, 128, 128, 0);

  // ---- fused link predictor ----
  link_predictor<<<EL / 16, 32, 0, stream>>>(z_inv, z_fund, eli_inv, eli_fund,
                                             Wp1, bp1, Wp2, bp2, (float*)d_out);
}